// GAT_62586263437627
// MI455X (gfx1250) — compile-verified
//
#include <hip/hip_runtime.h>
#include <math.h>

// GAT two-layer forward for MI455X (gfx1250), wave32.
// GEMMs use V_WMMA_F32_16X16X4_F32 (exact f32 numerics; GEMMs are not the
// bottleneck -- edge gather/scatter is, so no precision downgrade needed).

typedef float v2f __attribute__((ext_vector_type(2)));
typedef float v8f __attribute__((ext_vector_type(8)));

__device__ __forceinline__ float lrelu(float x) { return x > 0.f ? x : 0.2f * x; }

// monotone float <-> ordered-uint mapping (for atomicMax on floats)
__device__ __forceinline__ unsigned f2ord(float f) {
  unsigned u = __float_as_uint(f);
  return ((int)u < 0) ? ~u : (u ^ 0x80000000u);
}
__device__ __forceinline__ float ord2f(unsigned k) {
  return __uint_as_float(((int)k < 0) ? (k ^ 0x80000000u) : ~k);
}

// ---------------------------------------------------------------------------
// WMMA GEMM: C[M,N] = A[M,K] * B[K,N], all f32, one 16x16 tile per wave.
// Uses V_WMMA_F32_16X16X4_F32 with K-loop of K/4 steps.
// A 16x4 layout: lanes 0-15 hold K=k0,k0+1 ; lanes 16-31 hold K=k0+2,k0+3
// B 4x16 layout: VGPR0 = row (k0 | k0+2), VGPR1 = row (k0+1 | k0+3), N=lane&15
// C/D 16x16:     VGPR r = row r (lanes 0-15) / row r+8 (lanes 16-31)
// ---------------------------------------------------------------------------
template <int K, int N>
__global__ __launch_bounds__(256) void gat_wmma_gemm(
    const float* __restrict__ A, const float* __restrict__ B,
    float* __restrict__ C, int M, int tilesTotal) {
  const int wave = threadIdx.x >> 5;
  const int lane = threadIdx.x & 31;
  const int t = blockIdx.x * 8 + wave;
  if (t >= tilesTotal) return;                 // wave-uniform exit (EXEC stays full)
  constexpr int NT = N / 16;
  const int m0 = (t / NT) * 16;
  const int n0 = (t % NT) * 16;
  const int l16 = lane & 15;
  const int kh = (lane >> 4) << 1;             // 0 or 2

  int arow = m0 + l16;
  if (arow >= M) arow = M - 1;                 // clamped rows are never stored
  const float* Ar = A + (long)arow * K;

  v8f acc = {};
  for (int k0 = 0; k0 < K; k0 += 4) {
    v2f a = *(const v2f*)(Ar + k0 + kh);       // 8B-aligned (k0%4==0, kh even)
    v2f b;
    b.x = B[(k0 + kh + 0) * N + n0 + l16];
    b.y = B[(k0 + kh + 1) * N + n0 + l16];
    acc = __builtin_amdgcn_wmma_f32_16x16x4_f32(
        /*neg_a=*/false, a, /*neg_b=*/false, b,
        /*c_mod=*/(short)0, acc, /*reuse_a=*/false, /*reuse_b=*/false);
  }

  const int rbase = (lane >> 4) << 3;
#pragma unroll
  for (int r = 0; r < 8; ++r) {
    int m = m0 + rbase + r;
    if (m < M) C[(long)m * N + n0 + l16] = acc[r];
  }
}

// el[n,h] = dot(feat[n,h,:], al[h,:]) ; er likewise. One thread per (n,h).
__global__ void gat_rowdots(const float* __restrict__ feat,
                            const float* __restrict__ al,
                            const float* __restrict__ ar,
                            float* __restrict__ el, float* __restrict__ er,
                            int n, int D) {
  int t = blockIdx.x * blockDim.x + threadIdx.x;
  if (t >= n * 8) return;
  int node = t >> 3, h = t & 7;
  const float* f = feat + ((long)node * 8 + h) * D;
  const float* a0 = al + h * D;
  const float* a1 = ar + h * D;
  float sl = 0.f, sr = 0.f;
  for (int d = 0; d < D; ++d) {
    float v = f[d];
    sl += v * a0[d];
    sr += v * a1[d];
  }
  el[t] = sl;
  er[t] = sr;
}

// pass 1: segment max of leaky_relu(el[src]+er[dst]) grouped by dst
__global__ void gat_edge_max(const float* __restrict__ el,
                             const float* __restrict__ er,
                             const int* __restrict__ src,
                             const int* __restrict__ dst,
                             unsigned* __restrict__ emax, int E) {
  int e = blockIdx.x * blockDim.x + threadIdx.x;
  if (e >= E) return;
  int s = src[e], d = dst[e];
#pragma unroll
  for (int h = 0; h < 8; ++h) {
    float v = lrelu(el[s * 8 + h] + er[d * 8 + h]);
    atomicMax(&emax[d * 8 + h], f2ord(v));
  }
}

// pass 2: w = exp(e - max); denom[dst,h] += w; acc[dst,h,:] += w * feat[src,h,:]
template <int D>
__global__ void gat_edge_acc(const float* __restrict__ feat,
                             const float* __restrict__ el,
                             const float* __restrict__ er,
                             const int* __restrict__ src,
                             const int* __restrict__ dst,
                             const unsigned* __restrict__ emax,
                             float* __restrict__ denom, float* __restrict__ acc,
                             long EH) {
  long t = blockIdx.x * (long)blockDim.x + threadIdx.x;
  if (t >= EH) return;
  int e = (int)(t >> 3), h = (int)(t & 7);
  int s = src[e], d = dst[e];
  float v = lrelu(el[s * 8 + h] + er[d * 8 + h]);
  float w = expf(v - ord2f(emax[d * 8 + h]));
  atomicAdd(&denom[d * 8 + h], w);
  const float* fs = feat + ((long)s * 8 + h) * D;
  float* od = acc + ((long)d * 8 + h) * D;
#pragma unroll
  for (int j = 0; j < D; ++j) atomicAdd(&od[j], w * fs[j]);
}

// layer-1 finalize (in place): h = elu(acc/denom + b1)
__global__ void gat_final1(float* __restrict__ acc,
                           const float* __restrict__ denom,
                           const float* __restrict__ b, int n) {
  int t = blockIdx.x * blockDim.x + threadIdx.x;
  if (t >= n * 64) return;
  int node = t >> 6, c = t & 63, h = c >> 3;
  float dn = denom[node * 8 + h];
  float v = (dn > 0.f) ? acc[t] / dn : 0.f;
  v += b[c];
  acc[t] = (v > 0.f) ? v : (expf(v) - 1.f);
}

// layer-2 finalize: out[n,d] = mean_h(acc[n,h,d]/denom[n,h] + b2[h,d])
__global__ void gat_final2(const float* __restrict__ acc,
                           const float* __restrict__ denom,
                           const float* __restrict__ b,
                           float* __restrict__ out, int n) {
  int t = blockIdx.x * blockDim.x + threadIdx.x;
  if (t >= n * 32) return;
  int node = t >> 5, d = t & 31;
  float s = 0.f;
#pragma unroll
  for (int h = 0; h < 8; ++h) {
    float dn = denom[node * 8 + h];
    float v = (dn > 0.f) ? acc[((long)node * 8 + h) * 32 + d] / dn : 0.f;
    s += v + b[h * 32 + d];
  }
  out[t] = s * 0.125f;
}

static inline int cdiv_i(long a, long b) { return (int)((a + b - 1) / b); }

extern "C" void kernel_launch(void* const* d_in, const int* in_sizes, int n_in,
                              void* d_out, int out_size, void* d_ws, size_t ws_size,
                              hipStream_t stream) {
  const float* node_feat = (const float*)d_in[0];
  const float* W1  = (const float*)d_in[1];
  const float* al1 = (const float*)d_in[2];
  const float* ar1 = (const float*)d_in[3];
  const float* b1  = (const float*)d_in[4];
  const float* W2  = (const float*)d_in[5];
  const float* al2 = (const float*)d_in[6];
  const float* ar2 = (const float*)d_in[7];
  const float* b2  = (const float*)d_in[8];
  const int* src = (const int*)d_in[9];
  const int* dst = (const int*)d_in[10];

  const int N = in_sizes[0] / 128;   // 50000
  const int E = in_sizes[9];         // 1600000

  // workspace layout (floats): ~N*672*4 bytes total (~134 MB)
  float* feat1 = (float*)d_ws;                     // [N,64]
  float* acc1  = feat1 + (long)N * 64;             // [N,64]  -> becomes h (layer-2 input)
  float* feat2 = acc1  + (long)N * 64;             // [N,256]
  float* acc2  = feat2 + (long)N * 256;            // [N,256]
  float* el    = acc2  + (long)N * 256;            // [N,8]
  float* er    = el    + (long)N * 8;              // [N,8]
  float* denom = er    + (long)N * 8;              // [N,8]
  unsigned* emax = (unsigned*)(denom + (long)N * 8); // [N,8] ordered keys

  const int TB = 256;

  // ---------------- layer 1 ----------------
  hipMemsetAsync(acc1, 0, (size_t)N * 64 * sizeof(float), stream);
  hipMemsetAsync(denom, 0, (size_t)N * 8 * sizeof(float), stream);
  hipMemsetAsync(emax, 0, (size_t)N * 8 * sizeof(unsigned), stream);
  {
    int tiles = cdiv_i(N, 16) * (64 / 16);
    gat_wmma_gemm<128, 64><<<cdiv_i(tiles, 8), TB, 0, stream>>>(
        node_feat, W1, feat1, N, tiles);
  }
  gat_rowdots<<<cdiv_i((long)N * 8, TB), TB, 0, stream>>>(feat1, al1, ar1, el, er, N, 8);
  gat_edge_max<<<cdiv_i(E, TB), TB, 0, stream>>>(el, er, src, dst, emax, E);
  gat_edge_acc<8><<<cdiv_i((long)E * 8, TB), TB, 0, stream>>>(
      feat1, el, er, src, dst, emax, denom, acc1, (long)E * 8);
  gat_final1<<<cdiv_i((long)N * 64, TB), TB, 0, stream>>>(acc1, denom, b1, N);

  // ---------------- layer 2 ----------------
  hipMemsetAsync(acc2, 0, (size_t)N * 256 * sizeof(float), stream);
  hipMemsetAsync(denom, 0, (size_t)N * 8 * sizeof(float), stream);
  hipMemsetAsync(emax, 0, (size_t)N * 8 * sizeof(unsigned), stream);
  {
    int tiles = cdiv_i(N, 16) * (256 / 16);
    gat_wmma_gemm<64, 256><<<cdiv_i(tiles, 8), TB, 0, stream>>>(
        acc1, W2, feat2, N, tiles);
  }
  gat_rowdots<<<cdiv_i((long)N * 8, TB), TB, 0, stream>>>(feat2, al2, ar2, el, er, N, 32);
  gat_edge_max<<<cdiv_i(E, TB), TB, 0, stream>>>(el, er, src, dst, emax, E);
  gat_edge_acc<32><<<cdiv_i((long)E * 8, TB), TB, 0, stream>>>(
      feat2, el, er, src, dst, emax, denom, acc2, (long)E * 8);
  gat_final2<<<cdiv_i((long)N * 32, TB), TB, 0, stream>>>(acc2, denom, b2, (float*)d_out, N);
}